// WassersteinLoss_58179626992159
// MI455X (gfx1250) — compile-verified
//
#include <hip/hip_runtime.h>

typedef __attribute__((ext_vector_type(2))) float v2f;
typedef __attribute__((ext_vector_type(8))) float v8f;

#define DIM 256
#define BS 32768
#define REG_F 0.1f
#define EPS_F 1e-6f
#define NUM_ITER 100
#define LABEL_OFFSET 11

#define WAVES_PER_WG 4
#define ROWS_PER_WAVE 16
#define ROWS_PER_WG (WAVES_PER_WG * ROWS_PER_WAVE)  // 64
#define NUM_WG (BS / ROWS_PER_WG)                   // 512
#define LSTRIDE (DIM + 2)                           // 258: stride-2 banks, no 16-way conflict

// ---------- prep: K = exp(-10M - 1), transposed K / K*M / K*logK, and row sums ----------
__global__ __launch_bounds__(DIM) void prep_kernel(
    const float* __restrict__ M, float* __restrict__ KT, float* __restrict__ KMT,
    float* __restrict__ KLT, float* __restrict__ rowsums) {
  int j = blockIdx.x;   // row of K
  int k = threadIdx.x;  // col of K
  float m  = M[j * DIM + k];
  float lk = fmaf(-10.0f, m, -1.0f);   // log K = M/(-reg) - 1
  float Kv  = expf(lk);
  float KMv = Kv * m;
  float KLv = Kv * lk;
  KT [k * DIM + j] = Kv;               // transposed: column c of K is contiguous
  KMT[k * DIM + j] = KMv;
  KLT[k * DIM + j] = KLv;
  __shared__ float s0[DIM], s1[DIM], s2[DIM];
  s0[k] = Kv; s1[k] = KMv; s2[k] = KLv;
  __syncthreads();
  for (int off = DIM / 2; off > 0; off >>= 1) {
    if (k < off) { s0[k] += s0[k + off]; s1[k] += s1[k + off]; s2[k] += s2[k + off]; }
    __syncthreads();
  }
  if (k == 0) {
    rowsums[j]           = s0[0];   // rowsum K
    rowsums[DIM + j]     = s1[0];   // rowsum K*M
    rowsums[2 * DIM + j] = s2[0];   // rowsum K*logK
  }
}

// ---------- main: per-row Sinkhorn fixed point + WMMA rowsum dots ----------
__global__ __launch_bounds__(WAVES_PER_WG * 32) void sinkhorn_kernel(
    const float* __restrict__ pred, const int* __restrict__ label,
    const float* __restrict__ KT, const float* __restrict__ KMT,
    const float* __restrict__ KLT, const float* __restrict__ rowsums,
    double* __restrict__ partials) {

  __shared__ float  uLds[WAVES_PER_WG][ROWS_PER_WAVE][LSTRIDE];  // clamped u rows
  __shared__ float  bmat[16][LSTRIDE];   // B-matrix columns: n=0 rsKM, n=1 rsK, n=2 rsKL
  __shared__ float  rowScal[ROWS_PER_WG][8];
  __shared__ float  dmat[WAVES_PER_WG][16][16];
  __shared__ double contribS[WAVES_PER_WG][ROWS_PER_WAVE];

  const int tid  = threadIdx.x;
  const int wave = tid >> 5;
  const int lane = tid & 31;

  for (int idx = tid; idx < 16 * LSTRIDE; idx += blockDim.x) {
    int n = idx / LSTRIDE, j = idx - n * LSTRIDE;
    float v = 0.0f;
    if (j < DIM) {
      if (n == 0)      v = rowsums[DIM + j];
      else if (n == 1) v = rowsums[j];
      else if (n == 2) v = rowsums[2 * DIM + j];
    }
    bmat[n][j] = v;
  }
  __syncthreads();

  const int rowBase = blockIdx.x * ROWS_PER_WG + wave * ROWS_PER_WAVE;

  // ---- Phase 1: one wave per row, 8 elements/lane, 100 fixed-point iterations ----
  for (int rr = 0; rr < ROWS_PER_WAVE; ++rr) {
    const int row = rowBase + rr;
    const int c   = label[row] - LABEL_OFFSET;
    const float* predRow = pred + (size_t)row * DIM;
    const float* Kc  = KT  + (size_t)c * DIM;
    const float* KMc = KMT + (size_t)c * DIM;
    const float* KLc = KLT + (size_t)c * DIM;

    float p[8], kc[8], rkc[8], u[8];
    #pragma unroll
    for (int e = 0; e < 8; ++e) {
      int j = lane * 8 + e;
      p[e]   = predRow[j];
      kc[e]  = Kc[j];
      rkc[e] = 1.0f / kc[e];
      u[e]   = 1.0f / (float)DIM;
    }

    float s = 0.0f;
    for (int it = 0; it < NUM_ITER; ++it) {
      float acc = 0.0f;
      #pragma unroll
      for (int e = 0; e < 8; ++e) acc = fmaf(u[e], kc[e], acc);
      #pragma unroll
      for (int off = 16; off > 0; off >>= 1) acc += __shfl_xor(acc, off);
      s = acc;                               // (u @ K)[row, c]  -> v = 1/s
      #pragma unroll
      for (int e = 0; e < 8; ++e) u[e] = p[e] * (s * rkc[e]);  // u = pred/(v*Kc)
    }

    float vv = 1.0f / s;
    if (fabsf(vv) < EPS_F) vv = EPS_F;

    float A = 0, C = 0, E = 0, G = 0, D = 0;
    #pragma unroll
    for (int e = 0; e < 8; ++e) {
      int j = lane * 8 + e;
      float uc = u[e];
      if (fabsf(uc) < EPS_F) uc = EPS_F;     // where(|u|<eps, eps, u)
      float ul = uc * logf(uc);
      A = fmaf(uc, KMc[j],   A);   // sum u * KM[:,c]
      C = fmaf(ul, kc[e],    C);   // sum ulogu * K[:,c]
      E = fmaf(uc, KLc[j],   E);   // sum u * KlogK[:,c]
      G = fmaf(uc, kc[e],    G);   // sum u * K[:,c]
      D = fmaf(ul, bmat[1][j], D); // sum ulogu * rowsumK
      uLds[wave][rr][j] = uc;
    }
    #pragma unroll
    for (int off = 16; off > 0; off >>= 1) {
      A += __shfl_xor(A, off); C += __shfl_xor(C, off);
      E += __shfl_xor(E, off); G += __shfl_xor(G, off);
      D += __shfl_xor(D, off);
    }
    if (lane == 0) {
      float* rs = rowScal[wave * ROWS_PER_WAVE + rr];
      rs[0] = vv; rs[1] = A; rs[2] = C; rs[3] = E; rs[4] = G; rs[5] = D;
    }
  }
  __syncthreads();

  // ---- Phase 2: WMMA chain  D[16x16] = U[16x256] x B[256x16]  (f32, K=4 per op) ----
  {
    v8f acc = {};
    const int m  = lane & 15;          // M (rows of U) for A-frag, N (cols) for B-frag
    const int ks = (lane >> 4) * 2;    // lanes 0-15: K={0,1}; lanes 16-31: K={2,3}
    for (int kk = 0; kk < DIM / 4; ++kk) {
      const int j = kk * 4 + ks;
      v2f a, b;
      a.x = uLds[wave][m][j]; a.y = uLds[wave][m][j + 1];
      b.x = bmat[m][j];       b.y = bmat[m][j + 1];
      acc = __builtin_amdgcn_wmma_f32_16x16x4_f32(false, a, false, b, (short)0,
                                                  acc, false, false);
    }
    const int mb = (lane >> 4) * 8;    // C/D layout: lane=N, VGPR g -> M=g(+8)
    #pragma unroll
    for (int g = 0; g < 8; ++g) dmat[wave][mb + g][lane & 15] = acc[g];
  }
  __syncthreads();

  // ---- Phase 3: assemble per-row loss using eps-dense v structure ----
  if (lane < ROWS_PER_WAVE) {
    const float* rs = rowScal[wave * ROWS_PER_WAVE + lane];
    float vv = rs[0], A = rs[1], C = rs[2], E = rs[3], G = rs[4], D = rs[5];
    float Bv = dmat[wave][lane][0];    // sum u * rowsumKM
    float Hv = dmat[wave][lane][1];    // sum u * rowsumK
    float Fv = dmat[wave][lane][2];    // sum u * rowsumKlogK
    float ELE = EPS_F * logf(EPS_F);
    float vlv = vv * logf(vv);
    float loss_i = EPS_F * Bv + (vv - EPS_F) * A;
    float e1     = EPS_F * D  + (vv - EPS_F) * C;
    float e3     = EPS_F * Fv + (vv - EPS_F) * E;
    float e2     = ELE * (Hv - G) + vlv * G;
    contribS[wave][lane] =
        (double)loss_i + (double)REG_F * ((double)e1 + (double)e2 + (double)e3);
  }
  __syncthreads();
  if (tid == 0) {
    double t = 0.0;
    for (int i = 0; i < ROWS_PER_WG; ++i) t += contribS[i >> 4][i & 15];
    partials[blockIdx.x] = t;
  }
}

// ---------- final deterministic reduction ----------
__global__ __launch_bounds__(256) void reduce_kernel(const double* __restrict__ partials,
                                                     float* __restrict__ out) {
  __shared__ double sd[256];
  int t = threadIdx.x;
  sd[t] = partials[t] + partials[t + 256];
  __syncthreads();
  for (int off = 128; off > 0; off >>= 1) {
    if (t < off) sd[t] += sd[t + off];
    __syncthreads();
  }
  if (t == 0) out[0] = (float)(sd[0] / (double)BS);
}

extern "C" void kernel_launch(void* const* d_in, const int* in_sizes, int n_in,
                              void* d_out, int out_size, void* d_ws, size_t ws_size,
                              hipStream_t stream) {
  (void)in_sizes; (void)n_in; (void)out_size; (void)ws_size;
  const float* pred  = (const float*)d_in[0];
  const float* M     = (const float*)d_in[1];
  const int*   label = (const int*)d_in[2];
  float* out = (float*)d_out;

  float* KT      = (float*)d_ws;            // 256KB
  float* KMT     = KT  + DIM * DIM;         // 256KB
  float* KLT     = KMT + DIM * DIM;         // 256KB
  float* rowsums = KLT + DIM * DIM;         // 3KB
  double* partials = (double*)(rowsums + 3 * DIM);  // 4KB, 8B-aligned offset

  prep_kernel<<<DIM, DIM, 0, stream>>>(M, KT, KMT, KLT, rowsums);
  sinkhorn_kernel<<<NUM_WG, WAVES_PER_WG * 32, 0, stream>>>(
      pred, label, KT, KMT, KLT, rowsums, partials);
  reduce_kernel<<<1, 256, 0, stream>>>(partials, out);
}